// SoftSphereMultiModel_39281770889341
// MI455X (gfx1250) — compile-verified
//
#include <hip/hip_runtime.h>

// ---------------------------------------------------------------------------
// SoftSphere multi-species energy+forces, MI455X (gfx1250, wave32).
//
// Torus-chord embedding turns the periodic min-image cutoff test into a
// Gram-matrix threshold test computed with V_WMMA_F32_16X16X4_F32:
//   q_c = (L/2pi) * (cos(2pi x_c / L), sin(2pi x_c / L))  (6-D, padded to 8)
//   |q|^2 == 3R^2 for every particle  (R = L/2pi)
//   chord_r2 = 6R^2 - 2 q_i.q_j <= r2_minimage   (sin x <= x per axis)
//   => r_minimage < CUTOFF  ==>  dot > 3R^2 - CUTOFF^2/2   (exact superset,
//      survivors bounded by r < (pi/2)*CUTOFF, ~1% of all pairs)
// Exact min-image + pow on survivors only; division-free inner loop
// (1/sigma and eps/alpha pretabulated, v_rsq_f32 instead of sqrt+div).
// ---------------------------------------------------------------------------

typedef __attribute__((ext_vector_type(2))) float v2f;
typedef __attribute__((ext_vector_type(8))) float v8f;

#define NPART   4096
#define NTILES  (NPART / 16)         // 256 row tiles
#define CUTOFF  1.5f
#define CUT2    (CUTOFF * CUTOFF)
#define TWO_PI  6.283185307179586f

// d_ws layout (floats): [0, NPART*8)  embedding table
//                       [NPART*8, NPART*8 + NTILES)  per-tile partial energy
#define WS_EMB  0
#define WS_EN   (NPART * 8)

// ---------------------------------------------------------------- embedding
__global__ void k_embed(const float* __restrict__ pos,
                        const float* __restrict__ cell,
                        float* __restrict__ ws) {
  int i = blockIdx.x * blockDim.x + threadIdx.x;
  if (i >= NPART) return;
  const float L = cell[0];           // cubic cell: L * I
  const float R = L / TWO_PI;
  const float w = TWO_PI / L;
  float e[8];
#pragma unroll
  for (int c = 0; c < 3; ++c) {
    float s, co;
    __sincosf(pos[3 * i + c] * w, &s, &co);
    e[2 * c]     = R * co;
    e[2 * c + 1] = R * s;
  }
  e[6] = 0.0f;
  e[7] = 0.0f;
#pragma unroll
  for (int k = 0; k < 8; ++k) ws[WS_EMB + 8 * i + k] = e[k];
}

// -------------------------------------------------------------- pair kernel
// 128 threads = 4 waves; wave w owns row tile (blockIdx.x*4 + w).
__global__ __launch_bounds__(128)
void k_pairs(const float* __restrict__ pos,
             const float* __restrict__ cell,
             const float* __restrict__ sigM,
             const float* __restrict__ epsM,
             const float* __restrict__ alpM,
             const int*   __restrict__ spec,
             const float* __restrict__ emb,
             float* __restrict__ ws_en,
             float* __restrict__ forces) {
  __shared__ float sInvSig[64];                    // 1/sigma
  __shared__ float sEps[64];                       // eps
  __shared__ float sAlp[64];                       // alpha
  __shared__ float sEpsA[64];                      // eps/alpha
  __shared__ float sPosI[4][16][3];                // row-tile positions
  __shared__ int   sSpI[4][16];                    // row-tile species
  __shared__ float sF[4][16][3];                   // row-tile force reduce

  const int tid  = threadIdx.x;
  const int w    = tid >> 5;
  const int lane = tid & 31;
  const int half = lane >> 4;        // 0: rows 0-7 / 1: rows 8-15 (C layout)
  const int col  = lane & 15;        // N index of this lane
  const int tile = blockIdx.x * 4 + w;
  const int iBase = tile * 16;

  if (tid < 64) {                    // one-time divisions, amortized to zero
    const float sg = sigM[tid];
    const float al = alpM[tid];
    const float ep = epsM[tid];
    sInvSig[tid] = 1.0f / sg;
    sEps[tid]    = ep;
    sAlp[tid]    = al;
    sEpsA[tid]   = ep / al;
  }
  if (lane < 16) {
    const int i = iBase + lane;
    sPosI[w][lane][0] = pos[3 * i + 0];
    sPosI[w][lane][1] = pos[3 * i + 1];
    sPosI[w][lane][2] = pos[3 * i + 2];
    sSpI[w][lane] = spec[i];
    sF[w][lane][0] = 0.0f; sF[w][lane][1] = 0.0f; sF[w][lane][2] = 0.0f;
  }
  __syncthreads();

  const float L    = cell[0];
  const float invL = 1.0f / L;
  const float R    = L / TWO_PI;
  const float thresh = 3.0f * R * R - 0.5f * CUT2 - 1e-3f; // slack for fp32

  // A operand: 16x4 f32 layout -> lane holds row M = lane&15,
  // K = 2*half + {0,1} in the register pair. Two K-blocks for K=8.
  const int ia = iBase + col;
  v2f a_lo, a_hi;
  a_lo[0] = emb[8 * ia + 2 * half + 0];
  a_lo[1] = emb[8 * ia + 2 * half + 1];
  a_hi[0] = emb[8 * ia + 4 + 2 * half + 0];
  a_hi[1] = emb[8 * ia + 4 + 2 * half + 1];

  float eAcc = 0.0f;
  float fAcc[8][3];
#pragma unroll
  for (int v = 0; v < 8; ++v) { fAcc[v][0] = 0.0f; fAcc[v][1] = 0.0f; fAcc[v][2] = 0.0f; }

  for (int jt = 0; jt < NTILES; ++jt) {
    const int j = jt * 16 + col;

    // B operand: 4x16, mirrored gather (lane holds col N = lane&15).
    v2f b_lo, b_hi;
    b_lo[0] = emb[8 * j + 2 * half + 0];
    b_lo[1] = emb[8 * j + 2 * half + 1];
    b_hi[0] = emb[8 * j + 4 + 2 * half + 0];
    b_hi[1] = emb[8 * j + 4 + 2 * half + 1];

    const float pjx = pos[3 * j + 0];
    const float pjy = pos[3 * j + 1];
    const float pjz = pos[3 * j + 2];
    const int   sj  = spec[j];

    // Gram tile: D = A(16x8) x B(8x16) via two chained 16x16x4 f32 WMMAs.
    v8f acc = {0.f, 0.f, 0.f, 0.f, 0.f, 0.f, 0.f, 0.f};
    acc = __builtin_amdgcn_wmma_f32_16x16x4_f32(false, a_lo, false, b_lo,
                                                (short)0, acc, false, false);
    acc = __builtin_amdgcn_wmma_f32_16x16x4_f32(false, a_hi, false, b_hi,
                                                (short)0, acc, false, false);

#pragma unroll
    for (int v = 0; v < 8; ++v) {
      if (acc[v] > thresh) {                    // conservative cutoff filter
        const int m = v + 8 * half;             // row within tile (C layout)
        const int i = iBase + m;
        if (i != j) {
          float dx = pjx - sPosI[w][m][0];
          float dy = pjy - sPosI[w][m][1];
          float dz = pjz - sPosI[w][m][2];
          dx -= L * __builtin_rintf(dx * invL); // min-image wrap
          dy -= L * __builtin_rintf(dy * invL);
          dz -= L * __builtin_rintf(dz * invL);
          const float r2 = dx * dx + dy * dy + dz * dz;
          if (r2 < CUT2 && r2 > 1e-12f) {
            const float rinv = __builtin_amdgcn_rsqf(r2);   // v_rsq_f32
            const float r    = r2 * rinv;
            const int   p    = sSpI[w][m] * 8 + sj;
            const float isg  = sInvSig[p];
            const float f    = __builtin_fmaf(-r, isg, 1.0f);
            if (f > 0.0f) {
              // v_log_f32 / v_exp_f32 (base-2) hardware transcendentals
              const float al   = sAlp[p];
              const float lg   = __builtin_amdgcn_logf(f);
              const float fam1 = __builtin_amdgcn_exp2f((al - 1.0f) * lg); // f^(a-1)
              eAcc += sEpsA[p] * (fam1 * f);                               // eps/a*f^a
              const float c = -sEps[p] * isg * rinv * fam1;  // (dE/dr)/r
              fAcc[v][0] += c * dx;
              fAcc[v][1] += c * dy;
              fAcc[v][2] += c * dz;
            }
          }
        }
      }
    }
  }

  // Reduce per-lane force partials across the 16 lanes of each row (LDS).
#pragma unroll
  for (int v = 0; v < 8; ++v) {
    const int m = v + 8 * half;
    atomicAdd(&sF[w][m][0], fAcc[v][0]);
    atomicAdd(&sF[w][m][1], fAcc[v][1]);
    atomicAdd(&sF[w][m][2], fAcc[v][2]);
  }

  // Wave32 energy reduction -> per-tile partial (deterministic).
#pragma unroll
  for (int mask = 16; mask >= 1; mask >>= 1)
    eAcc += __shfl_xor(eAcc, mask, 32);
  if (lane == 0) ws_en[tile] = eAcc;

  __syncthreads();
  if (lane < 16) {                              // each row owned by one wave
    const int i = iBase + lane;
    forces[3 * i + 0] = sF[w][lane][0];
    forces[3 * i + 1] = sF[w][lane][1];
    forces[3 * i + 2] = sF[w][lane][2];
  }
}

// ----------------------------------------------------------- energy reduce
__global__ void k_final(const float* __restrict__ ws_en,
                        float* __restrict__ out_energy) {
  __shared__ float s[NTILES];
  const int t = threadIdx.x;
  s[t] = ws_en[t];
  __syncthreads();
  for (int o = NTILES / 2; o > 0; o >>= 1) {
    if (t < o) s[t] += s[t + o];
    __syncthreads();
  }
  if (t == 0) out_energy[0] = 0.5f * s[0];      // ordered-pair sum / 2
}

// ---------------------------------------------------------------- launcher
extern "C" void kernel_launch(void* const* d_in, const int* in_sizes, int n_in,
                              void* d_out, int out_size, void* d_ws, size_t ws_size,
                              hipStream_t stream) {
  const float* pos  = (const float*)d_in[0];
  const float* cell = (const float*)d_in[1];
  const float* sig  = (const float*)d_in[2];
  const float* eps  = (const float*)d_in[3];
  const float* alp  = (const float*)d_in[4];
  const int*   spc  = (const int*)d_in[5];

  float* out    = (float*)d_out;        // [energy, forces(4096*3)]
  float* ws     = (float*)d_ws;         // needs (4096*8 + 256)*4 = 132 KB
  float* ws_emb = ws + WS_EMB;
  float* ws_en  = ws + WS_EN;

  k_embed<<<NPART / 256, 256, 0, stream>>>(pos, cell, ws);
  k_pairs<<<NTILES / 4, 128, 0, stream>>>(pos, cell, sig, eps, alp, spc,
                                          ws_emb, ws_en, out + 1);
  k_final<<<1, NTILES, 0, stream>>>(ws_en, out);
}